// Graph_Readout_73151882985859
// MI455X (gfx1250) — compile-verified
//
#include <hip/hip_runtime.h>
#include <math.h>

// Graph readout (Set2Set): B=512 graphs, D=256, STEP=4.
// fp32 WMMA (V_WMMA_F32_16X16X4_F32) for the LSTM & output GEMMs,
// VALU + wave32 shuffles for the L2-resident segmented softmax attention.

typedef __attribute__((ext_vector_type(2))) float v2f;
typedef __attribute__((ext_vector_type(8))) float v8f;

constexpr int D     = 256;     // d_model
constexpr int D2    = 2 * D;   // 512
constexpr int D4    = 4 * D;   // 1024
constexpr int STEPS = 4;

__device__ __forceinline__ float sigf(float x) { return 1.0f / (1.0f + __expf(-x)); }

// ---------------------------------------------------------------------------
// One 16x16 fp32 C-tile accumulation: C += A[m0:m0+16, :ktot] * W[n0:n0+16, :ktot]^T
// A row-major [M, lda]; W row-major [N, ldw] (i.e. B = W^T, as in y = x @ W.T).
// ISA layout (CDNA5 ISA):
//   A (16x4 f32, 2 VGPRs): lane<16 -> M=lane, K=k0..k0+1; lane>=16 -> M=lane-16, K=k0+2..k0+3
//   B (4x16 f32, 2 VGPRs): lane<16 -> N=lane, K=k0..k0+1; lane>=16 -> N=lane-16, K=k0+2..k0+3
//   C/D (8 VGPRs): VGPR v -> M = v + 8*(lane>=16), N = lane%16
// ---------------------------------------------------------------------------
__device__ __forceinline__ void gemm_acc_k(const float* __restrict__ A, int lda,
                                           const float* __restrict__ W, int ldw,
                                           int ktot, int m0, int n0, int lane, v8f& acc) {
  const int half = lane >> 4;           // 0 or 1
  const int ksel = half * 2;            // K sub-select: 0 or 2
  const float* ap = A + (size_t)(m0 + (lane & 15)) * lda + ksel;
  const float* wp = W + (size_t)(n0 + (lane & 15)) * ldw + ksel;
  for (int k = 0; k < ktot; k += 4) {
    v2f a; a.x = ap[k]; a.y = ap[k + 1];
    v2f b; b.x = wp[k]; b.y = wp[k + 1];
    // (neg_a, A, neg_b, B, c_mod, C, reuse_a, reuse_b)
    acc = __builtin_amdgcn_wmma_f32_16x16x4_f32(false, a, false, b, (short)0, acc,
                                                false, false);
  }
}

// ---------------------------------------------------------------------------
// Kernel 0: segment start offsets (prefix sum of node_num). Trivial work.
// ---------------------------------------------------------------------------
__global__ void offs_kernel(const int* __restrict__ node_num, int* __restrict__ offs, int bsz) {
  if (blockIdx.x == 0 && threadIdx.x == 0) {
    int s = 0;
    for (int i = 0; i < bsz; ++i) { offs[i] = s; s += node_num[i]; }
  }
}

// ---------------------------------------------------------------------------
// Kernel 1: gates = q_star @ W_ih^T + h @ W_hh^T + b_ih + b_hh   -> [bsz, 4D]
// One wave per 16x16 output tile.  tiles: (bsz/16) x 64
// ---------------------------------------------------------------------------
__global__ void gates_kernel(const float* __restrict__ q_star, const float* __restrict__ h,
                             const float* __restrict__ W_ih, const float* __restrict__ W_hh,
                             const float* __restrict__ b_ih, const float* __restrict__ b_hh,
                             float* __restrict__ gates) {
  const int wave = (int)((blockIdx.x * blockDim.x + threadIdx.x) >> 5);
  const int lane = threadIdx.x & 31;
  const int nt = wave & (D4 / 16 - 1);  // 64 tiles along gate dim
  const int mt = wave >> 6;
  const int m0 = mt * 16, n0 = nt * 16;
  v8f acc = {};
  gemm_acc_k(q_star, D2, W_ih, D2, D2, m0, n0, lane, acc);  // K = 512
  gemm_acc_k(h,      D,  W_hh, D,  D,  m0, n0, lane, acc);  // K = 256
  const int nc = n0 + (lane & 15);
  const float bias = b_ih[nc] + b_hh[nc];
  const int mbase = m0 + 8 * (lane >> 4);
#pragma unroll
  for (int v = 0; v < 8; ++v)
    gates[(size_t)(mbase + v) * D4 + nc] = acc[v] + bias;
}

// ---------------------------------------------------------------------------
// Kernel 2: pointwise LSTM cell update (torch gate order i, f, g, o).
// Writes h, c and the q-half of q_star.
// ---------------------------------------------------------------------------
__global__ void lstm_pw_kernel(const float* __restrict__ gates, float* __restrict__ h,
                               float* __restrict__ c, float* __restrict__ q_star) {
  const int idx = (int)(blockIdx.x * blockDim.x + threadIdx.x);  // bsz*D threads
  const int b = idx >> 8;        // / D
  const int d = idx & (D - 1);
  const float* g = gates + (size_t)b * D4;
  const float gi = g[d], gf = g[D + d], gg = g[2 * D + d], go = g[3 * D + d];
  const float cv = sigf(gf) * c[idx] + sigf(gi) * tanhf(gg);
  const float hv = sigf(go) * tanhf(cv);
  c[idx] = cv;
  h[idx] = hv;
  q_star[(size_t)b * D2 + d] = hv;
}

// ---------------------------------------------------------------------------
// Kernel 3: segmented softmax attention, one block (8 waves) per graph.
//   e[n] = <node[n], q>;  a = softmax(e);  r = sum_n a[n]*node[n]
// q_star[b, D:2D] = r.   Node sweeps are coalesced and L2-resident (~105 MB).
// ---------------------------------------------------------------------------
__global__ void attn_kernel(const float* __restrict__ node, const int* __restrict__ node_num,
                            const int* __restrict__ offs, const float* __restrict__ h,
                            float* __restrict__ q_star, float* __restrict__ e_ws) {
  __shared__ float qs[D];
  __shared__ float red[8];
  __shared__ float bcast[2];
  const int b    = blockIdx.x;
  const int tid  = threadIdx.x;
  const int lane = tid & 31;
  const int wave = tid >> 5;
  const int base = offs[b];
  const int cnt  = node_num[b];

  qs[tid] = h[(size_t)b * D + tid];
  __syncthreads();

  // Phase 1: per-wave node dot products + running max.
  float wmax = -3.402823466e38f;
  for (int n = wave; n < cnt; n += 8) {
    const float* np = node + (size_t)(base + n) * D + lane * 8;
    const float* qp = qs + lane * 8;
    float s = 0.f;
#pragma unroll
    for (int j = 0; j < 8; ++j) s = fmaf(np[j], qp[j], s);
#pragma unroll
    for (int off = 16; off > 0; off >>= 1) s += __shfl_down(s, off, 32);
    s = __shfl(s, 0, 32);
    if (lane == 0) e_ws[base + n] = s;
    wmax = fmaxf(wmax, s);
  }
  if (lane == 0) red[wave] = wmax;
  __threadfence_block();
  __syncthreads();
  if (tid == 0) {
    float m = red[0];
#pragma unroll
    for (int j = 1; j < 8; ++j) m = fmaxf(m, red[j]);
    bcast[0] = m;
  }
  __syncthreads();
  const float emax = bcast[0];

  // Phase 2: exponentiate in place, block-wide sum.
  float lsum = 0.f;
  for (int t = tid; t < cnt; t += 256) {
    const float v = __expf(e_ws[base + t] - emax);
    e_ws[base + t] = v;
    lsum += v;
  }
#pragma unroll
  for (int off = 16; off > 0; off >>= 1) lsum += __shfl_down(lsum, off, 32);
  __syncthreads();                 // red[] reuse
  if (lane == 0) red[wave] = lsum;
  __threadfence_block();
  __syncthreads();
  if (tid == 0) {
    float t = 0.f;
#pragma unroll
    for (int j = 0; j < 8; ++j) t += red[j];
    bcast[1] = 1.0f / (t + 1e-6f);
  }
  __syncthreads();
  const float inv = bcast[1];

  // Phase 3: weighted sum, thread-per-dimension (coalesced row reads).
  float acc = 0.f;
  for (int n = 0; n < cnt; ++n)
    acc = fmaf(e_ws[base + n], node[(size_t)(base + n) * D + tid], acc);
  q_star[(size_t)b * D2 + D + tid] = acc * inv;
}

// ---------------------------------------------------------------------------
// Kernel 4: out = q_star @ Wo^T + bo   -> [bsz, D].  One wave per 16x16 tile.
// ---------------------------------------------------------------------------
__global__ void out_kernel(const float* __restrict__ q_star, const float* __restrict__ Wo,
                           const float* __restrict__ bo, float* __restrict__ out) {
  const int wave = (int)((blockIdx.x * blockDim.x + threadIdx.x) >> 5);
  const int lane = threadIdx.x & 31;
  const int nt = wave & (D / 16 - 1);   // 16 tiles along D
  const int mt = wave >> 4;
  const int m0 = mt * 16, n0 = nt * 16;
  v8f acc = {};
  gemm_acc_k(q_star, D2, Wo, D2, D2, m0, n0, lane, acc);   // K = 512
  const int nc = n0 + (lane & 15);
  const float bias = bo[nc];
  const int mbase = m0 + 8 * (lane >> 4);
#pragma unroll
  for (int v = 0; v < 8; ++v)
    out[(size_t)(mbase + v) * D + nc] = acc[v] + bias;
}

// ---------------------------------------------------------------------------
// Host launcher.  Inputs: node, node_num, W_ih, W_hh, b_ih, b_hh, Wo, bo.
// ---------------------------------------------------------------------------
extern "C" void kernel_launch(void* const* d_in, const int* in_sizes, int n_in,
                              void* d_out, int out_size, void* d_ws, size_t ws_size,
                              hipStream_t stream) {
  const float* node     = (const float*)d_in[0];
  const int*   node_num = (const int*)d_in[1];
  const float* W_ih     = (const float*)d_in[2];
  const float* W_hh     = (const float*)d_in[3];
  const float* b_ih     = (const float*)d_in[4];
  const float* b_hh     = (const float*)d_in[5];
  const float* Wo       = (const float*)d_in[6];
  const float* bo       = (const float*)d_in[7];
  float* out = (float*)d_out;

  const int bsz     = in_sizes[1];          // 512
  const int n_total = in_sizes[0] / D;      // 102400

  // Workspace carve-up (256 B aligned); ~4.6 MB total for the reference shape.
  char* w = (char*)d_ws;
  auto take = [&](size_t bytes) {
    char* p = w;
    w += (bytes + 255) & ~(size_t)255;
    return p;
  };
  int*   offs  = (int*)take((size_t)bsz * sizeof(int));
  float* gates = (float*)take((size_t)bsz * D4 * sizeof(float));
  float* hbuf  = (float*)take((size_t)bsz * D * sizeof(float));
  float* cbuf  = (float*)take((size_t)bsz * D * sizeof(float));
  float* qstar = (float*)take((size_t)bsz * D2 * sizeof(float));
  float* e_ws  = (float*)take((size_t)n_total * sizeof(float));

  // Deterministic init each call (h = c = q_star = 0).
  hipMemsetAsync(hbuf,  0, (size_t)bsz * D * sizeof(float), stream);
  hipMemsetAsync(cbuf,  0, (size_t)bsz * D * sizeof(float), stream);
  hipMemsetAsync(qstar, 0, (size_t)bsz * D2 * sizeof(float), stream);

  offs_kernel<<<1, 1, 0, stream>>>(node_num, offs, bsz);

  const int gate_waves  = (bsz / 16) * (D4 / 16);   // 2048
  const int gate_blocks = gate_waves / 8;           // 8 waves (256 thr) / block
  const int out_waves   = (bsz / 16) * (D / 16);    // 512
  const int out_blocks  = out_waves / 8;

  for (int s = 0; s < STEPS; ++s) {
    gates_kernel<<<gate_blocks, 256, 0, stream>>>(qstar, hbuf, W_ih, W_hh, b_ih, b_hh, gates);
    lstm_pw_kernel<<<bsz, 256, 0, stream>>>(gates, hbuf, cbuf, qstar);
    attn_kernel<<<bsz, 256, 0, stream>>>(node, node_num, offs, hbuf, qstar, e_ws);
  }
  out_kernel<<<out_blocks, 256, 0, stream>>>(qstar, Wo, bo, out);
}